// LowRankExperts_64656437674124
// MI455X (gfx1250) — compile-verified
//
#include <hip/hip_runtime.h>
#include <math.h>

// Problem constants (match reference)
#define E_  16
#define B_  1024
#define I_  512
#define O_  512
#define R_  8
#define H_  16
#define F_  8192                       // I*R + R*O
#define AHALF 4096                     // I*R  (first half of each f row)
#define A_TOTAL (67108864LL)           // E*B*AHALF elements in A region of d_out

typedef __attribute__((ext_vector_type(2))) float v2f;
typedef __attribute__((ext_vector_type(8))) float v8f;

__device__ __forceinline__ float fast_tanh(float v) {
#if __has_builtin(__builtin_amdgcn_tanhf)
    return __builtin_amdgcn_tanhf(v);       // gfx1250 V_TANH_F32 (TRANS op)
#else
    return tanhf(v);
#endif
}

// -------------------------------------------------------------------------
// Stage 1: h[e, b, :] = tanh(x[b, :] @ W1[e] + b1[e]),  h -> d_ws (1 MB)
// One wave computes one 16x16 tile of h (16 batch rows x 16 hidden) with a
// K-loop of 128 x v_wmma_f32_16x16x4_f32.
//
// A-frag (16x4 f32): lane L<16 holds x[m+L][k0], x[m+L][k0+1] (VGPR0,1);
//                    lane L>=16 holds K=k0+2, k0+3  -> one b64 load per lane.
// B-frag (4x16 f32): VGPR0 = row k0+2*hi, VGPR1 = row k0+1+2*hi, col = lane%16.
// -------------------------------------------------------------------------
__global__ __launch_bounds__(256)
void lre_stage1(const float* __restrict__ x,
                const float* __restrict__ W1,
                const float* __restrict__ bias1,
                float* __restrict__ hbuf)
{
    const int lane = threadIdx.x & 31;
    const int wave = threadIdx.x >> 5;
    const int l15  = lane & 15;
    const int hi   = lane >> 4;

    const int e  = blockIdx.y;
    const int m  = (blockIdx.x * 8 + wave) * 16;   // batch-row tile

    const float* xrow = x  + (size_t)(m + l15) * I_;
    const float* w1e  = W1 + (size_t)e * I_ * H_;

    v8f acc = {0.f, 0.f, 0.f, 0.f, 0.f, 0.f, 0.f, 0.f};

#pragma unroll 4
    for (int k0 = 0; k0 < I_; k0 += 4) {
        v2f a = *(const v2f*)(xrow + k0 + 2 * hi);          // b64 load
        v2f b;
        b.x = w1e[(size_t)(k0 + 2 * hi)     * H_ + l15];
        b.y = w1e[(size_t)(k0 + 2 * hi + 1) * H_ + l15];
        acc = __builtin_amdgcn_wmma_f32_16x16x4_f32(
                  false, a, false, b, (short)0, acc, false, false);
    }

    const float bv = bias1[e * H_ + l15];
    float* hp = hbuf + (size_t)(e * B_ + m + 8 * hi) * H_ + l15;
#pragma unroll
    for (int r = 0; r < 8; ++r) {
        hp[(size_t)r * H_] = fast_tanh(acc[r] + bv);        // row m + r + 8*hi
    }
}

// -------------------------------------------------------------------------
// Stage 2: f[e, b, :] = h[e, b, :] @ W2[e] + b2[e], scattered into the two
// output regions (A = f[:, :, :4096], Bf = f[:, :, 4096:]).
// One wave: one 16-row batch tile x 128 columns (8 N-tiles), K = 16 -> 4
// WMMAs per N-tile; A-frags of h are loaded once and reused across N-tiles.
// Output stores are non-temporal (536 MB streamed once, never re-read),
// keeping W2 (8 MB) and h (1 MB) resident in the 192 MB L2.
// -------------------------------------------------------------------------
__global__ __launch_bounds__(128)
void lre_stage2(const float* __restrict__ hbuf,
                const float* __restrict__ W2,
                const float* __restrict__ bias2,
                float* __restrict__ out)
{
    const int lane = threadIdx.x & 31;
    const int wave = threadIdx.x >> 5;
    const int l15  = lane & 15;
    const int hi   = lane >> 4;

    const int e      = blockIdx.z;
    const int m      = blockIdx.y * 16;
    const int n_base = (blockIdx.x * 4 + wave) * 128;

    // A-frags: h[e, m + l15, 0..15], one b64 load each (K pairs per ISA layout)
    const float* hrow = hbuf + (size_t)(e * B_ + m + l15) * H_;
    const v2f a0 = *(const v2f*)(hrow + 0  + 2 * hi);
    const v2f a1 = *(const v2f*)(hrow + 4  + 2 * hi);
    const v2f a2 = *(const v2f*)(hrow + 8  + 2 * hi);
    const v2f a3 = *(const v2f*)(hrow + 12 + 2 * hi);

    const float* w2e = W2 + (size_t)e * H_ * F_;
    const long long rowbase = (long long)(e * B_ + m + 8 * hi);

#pragma unroll
    for (int t = 0; t < 8; ++t) {
        const int n   = n_base + t * 16;
        const int col = n + l15;

        // B-frags: W2[e][k][col]; frag j covers K = 4j..4j+3.
        const float* wp = w2e + (size_t)(2 * hi) * F_ + col;
        v2f b0, b1, b2, b3;
        b0.x = wp[0 * F_];   b0.y = wp[1 * F_];
        b1.x = wp[4 * F_];   b1.y = wp[5 * F_];
        b2.x = wp[8 * F_];   b2.y = wp[9 * F_];
        b3.x = wp[12 * F_];  b3.y = wp[13 * F_];

        v8f acc = {0.f, 0.f, 0.f, 0.f, 0.f, 0.f, 0.f, 0.f};
        acc = __builtin_amdgcn_wmma_f32_16x16x4_f32(false, a0, false, b0, (short)0, acc, false, false);
        acc = __builtin_amdgcn_wmma_f32_16x16x4_f32(false, a1, false, b1, (short)0, acc, false, false);
        acc = __builtin_amdgcn_wmma_f32_16x16x4_f32(false, a2, false, b2, (short)0, acc, false, false);
        acc = __builtin_amdgcn_wmma_f32_16x16x4_f32(false, a3, false, b3, (short)0, acc, false, false);

        const float bv = bias2[(size_t)e * F_ + col];

        // Column block lies entirely in one output region (4096 % 16 == 0).
        long long off = (col < AHALF)
                      ? rowbase * (long long)AHALF + col
                      : A_TOTAL + rowbase * (long long)AHALF + (col - AHALF);
        float* p = out + off;
#pragma unroll
        for (int r = 0; r < 8; ++r) {
            __builtin_nontemporal_store(acc[r] + bv, p + (size_t)r * AHALF);
        }
    }
}

// -------------------------------------------------------------------------
// d_in order: x[B,I], W1[E,I,H], b1[E,H], W2[E,H,F], b2[E,F]  (all fp32)
// d_out: A[E,B,I,R] flat (67108864 floats) then Bf[E,B,R,O] flat.
// d_ws: stage-1 activations h[E,B,H] = 1 MB fp32.
// -------------------------------------------------------------------------
extern "C" void kernel_launch(void* const* d_in, const int* in_sizes, int n_in,
                              void* d_out, int out_size, void* d_ws, size_t ws_size,
                              hipStream_t stream)
{
    const float* x  = (const float*)d_in[0];
    const float* W1 = (const float*)d_in[1];
    const float* b1 = (const float*)d_in[2];
    const float* W2 = (const float*)d_in[3];
    const float* b2 = (const float*)d_in[4];
    float* out  = (float*)d_out;
    float* hbuf = (float*)d_ws;                 // E*B*H floats = 1 MB

    // Stage 1: grid = (B/16/8 m-tiles, E), 8 waves per block, 1 tile per wave.
    lre_stage1<<<dim3(B_ / 16 / 8, E_), 256, 0, stream>>>(x, W1, b1, hbuf);

    // Stage 2: grid = (F/512 col-groups, B/16 m-tiles, E), 4 waves per block.
    lre_stage2<<<dim3(F_ / 512, B_ / 16, E_), 128, 0, stream>>>(hbuf, W2, b2, out);
}